// SpGCN_73212012527836
// MI455X (gfx1250) — compile-verified
//
#include <hip/hip_runtime.h>

typedef __attribute__((ext_vector_type(2))) float v2f;
typedef __attribute__((ext_vector_type(8))) float v8f;

#define FDIM 128
#define BN_EPS 1e-5f

// ---------------------------------------------------------------------------
// GEMM: C[M x 128] = A[M x 128] @ W[128 x 128] using V_WMMA_F32_16X16X4_F32.
// Block = 256 threads = 8 waves. Each block owns a 16-row slab; wave w owns
// the 16x16 output tile at columns [16w, 16w+16). K loop in steps of 4.
//
// fp32 WMMA fragment layout (ISA 7.12.2):
//   A (16x4): lanes 0-15 -> M=lane, VGPR0=K0 VGPR1=K1; lanes 16-31 -> K2/K3
//   B (4x16): VGPR0 = rows K0 (lanes 0-15) / K2 (lanes 16-31); VGPR1 = K1/K3
//   C/D (16x16): VGPR r = row r (lanes 0-15) / row r+8 (lanes 16-31)
// ---------------------------------------------------------------------------
__global__ __launch_bounds__(256) void spgcn_gemm_wmma_f32(
    const float* __restrict__ A, const float* __restrict__ W,
    float* __restrict__ C)
{
    const int wave  = threadIdx.x >> 5;
    const int lane  = threadIdx.x & 31;
    const int lm    = lane & 15;          // row within tile (A) / col within tile (B)
    const int khalf = (lane >> 4) << 1;   // 0 for lanes 0-15, 2 for lanes 16-31
    const int m_base = blockIdx.x * 16;
    const int n_base = wave * 16;

    const float* __restrict__ arow = A + (size_t)(m_base + lm) * FDIM;

    v8f acc = {0.f, 0.f, 0.f, 0.f, 0.f, 0.f, 0.f, 0.f};

    #pragma unroll 4
    for (int k = 0; k < FDIM; k += 4) {
        v2f a;
        a.x = arow[k + khalf + 0];
        a.y = arow[k + khalf + 1];
        v2f b;
        b.x = W[(size_t)(k + khalf + 0) * FDIM + n_base + lm];
        b.y = W[(size_t)(k + khalf + 1) * FDIM + n_base + lm];
        acc = __builtin_amdgcn_wmma_f32_16x16x4_f32(
            /*neg_a=*/false, a, /*neg_b=*/false, b,
            /*c_mod=*/(short)0, acc, /*reuse_a=*/false, /*reuse_b=*/false);
    }

    const int m_off = (lane >> 4) << 3; // 0 or 8
    #pragma unroll
    for (int r = 0; r < 8; ++r) {
        C[(size_t)(m_base + m_off + r) * FDIM + n_base + lm] = acc[r];
    }
}

// ---------------------------------------------------------------------------
// Initialize an [N x 128] buffer with a broadcast bias vector (float4 wide).
// ---------------------------------------------------------------------------
__global__ __launch_bounds__(256) void spgcn_bias_init(
    float* __restrict__ out, const float* __restrict__ bias, long long total4)
{
    long long t = (long long)blockIdx.x * blockDim.x + threadIdx.x;
    if (t >= total4) return;
    long long i = t * 4;
    const int f = (int)(i & (FDIM - 1));
    float4 b = *(const float4*)(bias + f);
    *(float4*)(out + i) = b;
}

// ---------------------------------------------------------------------------
// SpMM (COO): out[row[e]] += w[e] * H[col[e]]. One wave per edge; each lane
// handles 4 consecutive features (float4 load, 4 global f32 atomics).
// Destinations are L2-resident (51 MB << 192 MB L2) so global_atomic_add_f32
// resolves in-cache.
// ---------------------------------------------------------------------------
__global__ __launch_bounds__(256) void spgcn_spmm_atomic(
    const float* __restrict__ H, const int* __restrict__ erow,
    const int* __restrict__ ecol, const float* __restrict__ ew,
    float* __restrict__ out, int E)
{
    long long gid = (long long)blockIdx.x * blockDim.x + threadIdx.x;
    int e = (int)(gid >> 5);
    if (e >= E) return;
    int c = ((int)gid & 31) << 2;   // feature offset: 0,4,...,124

    const int src = ecol[e];
    const int dst = erow[e];
    const float wt = ew[e];

    float4 v = *(const float4*)(H + (size_t)src * FDIM + c);
    float* o = out + (size_t)dst * FDIM + c;
    unsafeAtomicAdd(o + 0, v.x * wt);
    unsafeAtomicAdd(o + 1, v.y * wt);
    unsafeAtomicAdd(o + 2, v.z * wt);
    unsafeAtomicAdd(o + 3, v.w * wt);
}

// ---------------------------------------------------------------------------
// BN stats pass: per-feature sum and sum-of-squares into stats[0..127] and
// stats[128..255]. Block covers 256 rows; thread (f, half) strides rows by 2.
// ---------------------------------------------------------------------------
__global__ __launch_bounds__(256) void spgcn_bn_stats(
    const float* __restrict__ H, float* __restrict__ stats, int N)
{
    const int f    = threadIdx.x & (FDIM - 1);
    const int half = threadIdx.x >> 7;              // 0 or 1
    const int r0   = blockIdx.x * 256 + half;

    float s = 0.f, s2 = 0.f;
    #pragma unroll 4
    for (int i = 0; i < 128; ++i) {
        int r = r0 + i * 2;
        if (r < N) {
            float v = H[(size_t)r * FDIM + f];
            s  += v;
            s2 += v * v;
        }
    }
    unsafeAtomicAdd(&stats[f], s);
    unsafeAtomicAdd(&stats[FDIM + f], s2);
}

// ---------------------------------------------------------------------------
// BN apply + ReLU, in place, float4 wide.
//   mean = sum/N ; var = sumsq/N - mean^2
//   h = relu(gamma * (h - mean) * rsqrt(var + eps) + beta)
// ---------------------------------------------------------------------------
__global__ __launch_bounds__(256) void spgcn_bn_apply_relu(
    float* __restrict__ H, const float* __restrict__ stats,
    const float* __restrict__ gamma, const float* __restrict__ beta,
    long long total4, float invN)
{
    long long t = (long long)blockIdx.x * blockDim.x + threadIdx.x;
    if (t >= total4) return;
    long long i = t * 4;
    const int f = (int)(i & (FDIM - 1));

    float4 h  = *(float4*)(H + i);
    float4 s  = *(const float4*)(stats + f);
    float4 s2 = *(const float4*)(stats + FDIM + f);
    float4 g  = *(const float4*)(gamma + f);
    float4 b  = *(const float4*)(beta + f);

    float m, var, sc;
    m = s.x * invN; var = fmaxf(s2.x * invN - m * m, 0.f);
    sc = g.x * __frsqrt_rn(var + BN_EPS);
    h.x = fmaxf((h.x - m) * sc + b.x, 0.f);

    m = s.y * invN; var = fmaxf(s2.y * invN - m * m, 0.f);
    sc = g.y * __frsqrt_rn(var + BN_EPS);
    h.y = fmaxf((h.y - m) * sc + b.y, 0.f);

    m = s.z * invN; var = fmaxf(s2.z * invN - m * m, 0.f);
    sc = g.z * __frsqrt_rn(var + BN_EPS);
    h.z = fmaxf((h.z - m) * sc + b.z, 0.f);

    m = s.w * invN; var = fmaxf(s2.w * invN - m * m, 0.f);
    sc = g.w * __frsqrt_rn(var + BN_EPS);
    h.w = fmaxf((h.w - m) * sc + b.w, 0.f);

    *(float4*)(H + i) = h;
}

// ---------------------------------------------------------------------------
// Orchestration
// ---------------------------------------------------------------------------
extern "C" void kernel_launch(void* const* d_in, const int* in_sizes, int n_in,
                              void* d_out, int out_size, void* d_ws, size_t ws_size,
                              hipStream_t stream) {
    const float* x     = (const float*)d_in[0];
    const int*   erow  = (const int*)d_in[1];
    const int*   ecol  = (const int*)d_in[2];
    const float* ew    = (const float*)d_in[3];
    const float* W1    = (const float*)d_in[4];
    const float* b1    = (const float*)d_in[5];
    const float* W2    = (const float*)d_in[6];
    const float* b2    = (const float*)d_in[7];
    const float* gamma = (const float*)d_in[8];
    const float* beta  = (const float*)d_in[9];

    const int N = in_sizes[0] / FDIM;   // 100000
    const int E = in_sizes[1];          // 1600000
    const long long NF  = (long long)N * FDIM;
    const long long NF4 = NF / 4;

    // Workspace layout: bufA | bufB | stats(256 floats)
    float* bufA  = (float*)d_ws;            // xW1, later h1@W2
    float* bufB  = bufA + NF;               // spmm1 result -> BN/ReLU in place
    float* stats = bufB + NF;               // [sum(128) | sumsq(128)]
    float* out   = (float*)d_out;

    const int mtiles      = N / 16;                       // 6250 (exact)
    const int ew_blocks   = (int)(((long long)E * 32 + 255) / 256);
    const int elem_blocks = (int)((NF4 + 255) / 256);
    const int stat_blocks = (N + 255) / 256;

    // Layer 1: h0 = x @ W1
    spgcn_gemm_wmma_f32<<<mtiles, 256, 0, stream>>>(x, W1, bufA);

    // h = spmm(h0) + b1  (init dest with b1, then atomic accumulate)
    spgcn_bias_init<<<elem_blocks, 256, 0, stream>>>(bufB, b1, NF4);
    spgcn_spmm_atomic<<<ew_blocks, 256, 0, stream>>>(bufA, erow, ecol, ew, bufB, E);

    // BatchNorm (batch stats) + ReLU in place on bufB
    hipMemsetAsync(stats, 0, 2 * FDIM * sizeof(float), stream);
    spgcn_bn_stats<<<stat_blocks, 256, 0, stream>>>(bufB, stats, N);
    spgcn_bn_apply_relu<<<elem_blocks, 256, 0, stream>>>(bufB, stats, gamma, beta,
                                                         NF4, 1.0f / (float)N);

    // Layer 2: h2 = h1 @ W2
    spgcn_gemm_wmma_f32<<<mtiles, 256, 0, stream>>>(bufB, W2, bufA);

    // out = spmm(h2) + b2
    spgcn_bias_init<<<elem_blocks, 256, 0, stream>>>(out, b2, NF4);
    spgcn_spmm_atomic<<<ew_blocks, 256, 0, stream>>>(bufA, erow, ecol, ew, out, E);
}